// GATLayer_68487548502750
// MI455X (gfx1250) — compile-verified
//
#include <hip/hip_runtime.h>

#define Bb 4
#define Hh 4
#define Ss 2048
#define Dd 256
#define ALPHA 0.2f

typedef __attribute__((ext_vector_type(2))) float v2f;
typedef __attribute__((ext_vector_type(8))) float v8f;
typedef int vi4 __attribute__((vector_size(16)));          // int4, GCC-vector style
typedef __attribute__((address_space(1))) vi4 gvi4;        // global int4
typedef __attribute__((address_space(3))) vi4 lvi4;        // LDS int4

static __device__ __forceinline__ v8f wmma_f32(v2f a, v2f b, v8f c) {
    return __builtin_amdgcn_wmma_f32_16x16x4_f32(false, a, false, b, (short)0, c,
                                                 false, false);
}

// ---- CDNA5 async global->LDS (ASYNCcnt-tracked DMA) -----------------------
static __device__ __forceinline__ void async_ld_b128(const float4* g, float4* l) {
#if __has_builtin(__builtin_amdgcn_global_load_async_to_lds_b128)
    __builtin_amdgcn_global_load_async_to_lds_b128(
        (gvi4*)g, (lvi4*)l, 0, 0);
#else
    // GV mode: vdst = LDS byte address (low 32 bits of generic shared ptr),
    // vaddr = 64-bit global address.
    asm volatile("global_load_async_to_lds_b128 %0, %1, off"
                 :: "v"((unsigned)(unsigned long long)l),
                    "v"((unsigned long long)g)
                 : "memory");
#endif
}

#if __has_builtin(__builtin_amdgcn_s_wait_asynccnt)
#define WAIT_ASYNC(n) __builtin_amdgcn_s_wait_asynccnt(n)
#else
#define WAIT_ASYNC(n) asm volatile("s_wait_asynccnt %0" :: "i"(n) : "memory")
#endif

// ---------------------------------------------------------------------------
// Kernel 1: h[b,h,s,e] = sum_d x[b,s,d] * W[h,e,d]   (16-row strip per block)
//           + s1[b,h,s] = h . a1,  s2[b,h,s] = h . a2
// ---------------------------------------------------------------------------
__global__ __launch_bounds__(256) void gat_h_scores(
    const float* __restrict__ x,   // [B,S,D]
    const float* __restrict__ W,   // [H,D,D]  (e-major, d inner)
    const float* __restrict__ a,   // [H,2D]
    float* __restrict__ hbuf,      // ws [B,H,S,D]
    float* __restrict__ s1g,       // ws [B,H,S]
    float* __restrict__ s2g)       // ws [B,H,S]
{
    __shared__ float xt[16 * Dd];  // 16KB: x tile, later reused as h tile

    const int tid  = threadIdx.x;
    const int lane = tid & 31;
    const int wave = tid >> 5;
    const int s0   = blockIdx.x * 16;
    const int bh   = blockIdx.y;
    const int b    = bh / Hh;
    const int hd   = bh % Hh;

    // async-stage x[b, s0:s0+16, :] -> LDS (coalesced b128, no VGPR round trip)
    {
        const float4* src4 = (const float4*)(x + ((size_t)b * Ss + s0) * Dd);
        float4* dst4 = (float4*)xt;
        for (int i = tid; i < 16 * Dd / 4; i += 256)   // 4 per thread, uniform
            async_ld_b128(src4 + i, dst4 + i);
    }
    WAIT_ASYNC(0);
    __syncthreads();

    const int nidx  = lane & 15;          // M for A-frag, N for B/C frags
    const int koff  = (lane >> 4) * 2;    // K sub-offset: 0 (lanes 0-15) / 2
    const int rbase = (lane >> 4) * 8;    // C/D row base
    const int e0    = wave * 32;          // this wave covers cols e0..e0+31

    const float* Wh = W + (size_t)hd * Dd * Dd;
    v8f c0 = {}; v8f c1 = {};

    for (int k = 0; k < Dd; k += 4) {
        v2f af;
        af[0] = xt[nidx * Dd + k + koff];
        af[1] = xt[nidx * Dd + k + koff + 1];
        const float* w0 = Wh + (size_t)(e0 + nidx) * Dd + k + koff;
        const float* w1 = Wh + (size_t)(e0 + 16 + nidx) * Dd + k + koff;
        v2f bf0, bf1;
        bf0[0] = w0[0]; bf0[1] = w0[1];
        bf1[0] = w1[0]; bf1[1] = w1[1];
        c0 = wmma_f32(af, bf0, c0);
        c1 = wmma_f32(af, bf1, c1);
    }

    __syncthreads();  // everyone done reading xt; reuse it as h tile

    float* hdst = hbuf + ((size_t)bh * Ss + s0) * Dd;
    for (int r = 0; r < 8; ++r) {
        const int m = r + rbase;
        hdst[(size_t)m * Dd + e0 + nidx]      = c0[r];
        hdst[(size_t)m * Dd + e0 + 16 + nidx] = c1[r];
        xt[m * Dd + e0 + nidx]      = c0[r];
        xt[m * Dd + e0 + 16 + nidx] = c1[r];
    }
    __syncthreads();

    // s1/s2 for the 16 rows of this strip (threads 0..31: 16 rows x {s1,s2})
    if (tid < 32) {
        const int p   = tid & 15;
        const int sel = tid >> 4;
        const float* av = a + (size_t)hd * (2 * Dd) + sel * Dd;
        float acc = 0.f;
        for (int e = 0; e < Dd; ++e) acc += xt[p * Dd + e] * av[e];
        (sel ? s2g : s1g)[(size_t)bh * Ss + s0 + p] = acc;
    }
}

// ---------------------------------------------------------------------------
// Kernel 2: flash-style attention + output, mean over heads.
//   e[p,q] = lrelu(s1[p]+s2[q]); attn = softmax_q(e); out = attn @ h; mean_h
// One block owns a 16-row p-tile of one batch, loops all H heads.
// h is double-buffered into LDS with async b128 copies issued one chunk ahead.
// ---------------------------------------------------------------------------
__global__ __launch_bounds__(256) void gat_attn(
    const float* __restrict__ hbuf,  // [B,H,S,D]
    const float* __restrict__ s1g,   // [B,H,S]
    const float* __restrict__ s2g,   // [B,H,S]
    float* __restrict__ out)         // [B,S,D]
{
    __shared__ float hs[2][32 * Dd]; // 2 x 32KB double-buffered h chunks
    __shared__ float P[16 * 32];     // exp'd probability chunk
    __shared__ float s2c[32];
    __shared__ float s1t[16];
    __shared__ float mrow[16], lrow[16], frow[16];

    const int tid  = threadIdx.x;
    const int lane = tid & 31;
    const int wave = tid >> 5;
    const int p0   = blockIdx.x * 16;
    const int b    = blockIdx.y;

    const int nidx  = lane & 15;
    const int koff  = (lane >> 4) * 2;
    const int rbase = (lane >> 4) * 8;
    const int e0    = wave * 32;

    v8f acc0 = {}; v8f acc1 = {};   // head-sum accumulators

    for (int hd = 0; hd < Hh; ++hd) {
        const int bh = b * Hh + hd;
        const float* hmat = hbuf + (size_t)bh * Ss * Dd;

        if (tid < 16) {
            s1t[tid]  = s1g[(size_t)bh * Ss + p0 + tid];
            mrow[tid] = -3.0e38f;
            lrow[tid] = 0.f;
            frow[tid] = 1.f;
        }

        // prefetch chunk 0 into buffer 0 (8 async b128 per thread)
        {
            const float4* g4 = (const float4*)hmat;
            float4* l4 = (float4*)hs[0];
            for (int i = tid; i < 32 * Dd / 4; i += 256)
                async_ld_b128(g4 + i, l4 + i);
        }
        __syncthreads();

        v8f c0 = {}; v8f c1 = {};

        for (int q0 = 0; q0 < Ss; q0 += 32) {
            const int cur = (q0 >> 5) & 1;
            const bool hasNext = (q0 + 32) < Ss;

            // issue async prefetch of the next chunk into the other buffer
            if (hasNext) {
                const float4* g4 = (const float4*)(hmat + (size_t)(q0 + 32) * Dd);
                float4* l4 = (float4*)hs[cur ^ 1];
                for (int i = tid; i < 32 * Dd / 4; i += 256)
                    async_ld_b128(g4 + i, l4 + i);
            }

            if (tid < 32) s2c[tid] = s2g[(size_t)bh * Ss + q0 + tid];
            __syncthreads();

            // raw leaky-relu logits: 512 values, 2 per thread (overlaps DMA)
            {
                float v0 = s1t[tid >> 5] + s2c[tid & 31];
                P[tid] = v0 > 0.f ? v0 : ALPHA * v0;
                const int i1 = tid + 256;
                float v1 = s1t[i1 >> 5] + s2c[i1 & 31];
                P[i1] = v1 > 0.f ? v1 : ALPHA * v1;
            }
            __syncthreads();

            // running softmax state per row (threads 0..15)
            if (tid < 16) {
                float m  = mrow[tid];
                float cm = m;
                for (int q = 0; q < 32; ++q) cm = fmaxf(cm, P[tid * 32 + q]);
                const float f = __expf(m - cm);
                float l = lrow[tid] * f;
                for (int q = 0; q < 32; ++q) {
                    const float pe = __expf(P[tid * 32 + q] - cm);
                    P[tid * 32 + q] = pe;
                    l += pe;
                }
                mrow[tid] = cm; lrow[tid] = l; frow[tid] = f;
            }

            // drain this wave's copies of the current buffer (next stays in
            // flight), then the barrier publishes every wave's LDS writes
            if (hasNext) { WAIT_ASYNC(8); } else { WAIT_ASYNC(0); }
            __syncthreads();

            // rescale accumulators by exp(m_old - m_new) for each row
            for (int r = 0; r < 8; ++r) {
                const float f = frow[r + rbase];
                c0[r] *= f; c1[r] *= f;
            }

            // C += P(16xK) @ h(KxN) over this 32-wide q-chunk, h from LDS
            const float* hc = hs[cur];
            for (int kk = 0; kk < 32; kk += 4) {
                v2f af;
                af[0] = P[nidx * 32 + kk + koff];
                af[1] = P[nidx * 32 + kk + koff + 1];
                const float* hb = hc + (kk + koff) * Dd + e0 + nidx;
                v2f bf0, bf1;
                bf0[0] = hb[0];  bf0[1] = hb[Dd];
                bf1[0] = hb[16]; bf1[1] = hb[Dd + 16];
                c0 = wmma_f32(af, bf0, c0);
                c1 = wmma_f32(af, bf1, c1);
            }
            __syncthreads();  // reads of hs[cur]/P done before reuse
        }

        // fold this head into the mean accumulator: acc += C / l[row]
        for (int r = 0; r < 8; ++r) {
            const float inv = 1.0f / lrow[r + rbase];
            acc0[r] += c0[r] * inv;
            acc1[r] += c1[r] * inv;
        }
        __syncthreads();  // before next head rewrites s1t/mrow/lrow & hs[0]
    }

    float* od = out + ((size_t)b * Ss + p0) * Dd;
    const float invH = 1.0f / (float)Hh;
    for (int r = 0; r < 8; ++r) {
        const int m = r + rbase;
        od[(size_t)m * Dd + e0 + nidx]      = acc0[r] * invH;
        od[(size_t)m * Dd + e0 + 16 + nidx] = acc1[r] * invH;
    }
}

// ---------------------------------------------------------------------------
extern "C" void kernel_launch(void* const* d_in, const int* in_sizes, int n_in,
                              void* d_out, int out_size, void* d_ws, size_t ws_size,
                              hipStream_t stream) {
    const float* x = (const float*)d_in[0];   // [4,2048,256]
    const float* W = (const float*)d_in[1];   // [4,256,256]
    const float* a = (const float*)d_in[2];   // [4,512]
    float* out = (float*)d_out;               // [4,2048,256]

    float* h  = (float*)d_ws;                          // 32 MB
    float* s1 = h  + (size_t)Bb * Hh * Ss * Dd;        // 128 KB
    float* s2 = s1 + (size_t)Bb * Hh * Ss;             // 128 KB

    dim3 g1(Ss / 16, Bb * Hh);
    gat_h_scores<<<g1, 256, 0, stream>>>(x, W, a, h, s1, s2);

    dim3 g2(Ss / 16, Bb);
    gat_attn<<<g2, 256, 0, stream>>>(h, s1, s2, out);
}